// MaSIF_ligand_88510686036217
// MI455X (gfx1250) — compile-verified
//
#include <hip/hip_runtime.h>

// MaSIF geodesic conv for gfx1250 (MI455X), fp32 WMMA path.
// Factorization: bins b = r*16 + t with mu_rho dep only on r, mu_theta only on t.
//   Q[v, 2r+0] = rho_g[v,r]*feat_i[v],  Q[v, 2r+1] = rho_g[v,r]   (k-independent)
//   P[v, t]    = th_g[v,t,k]*mask[v]
//   D = Q^T P  (16x16 via V_WMMA_F32_16X16X4_F32, K=128)  -> desc = num/(den+eps)
//   Conv: Desc(16 rot x 80) @ W_i(80x80) + b_i  via WMMA (M=16,K=80,N=80), max rows, relu.

#define NV 128
#define NF 4
#define NT 16
#define NR 5
#define NB 80
#define NROT 16
#define TWO_PI_F 6.28318530717958647692f
#define ROT_STEP (TWO_PI_F / 16.0f)
#define EPSF 1e-5f

typedef float v2f __attribute__((ext_vector_type(2)));
typedef float v8f __attribute__((ext_vector_type(8)));

#define QSTRIDE 130              // 128 + 2 pad -> conflict-free b64 row reads
#define DSTRIDE 82               // 80 + 2 pad
#define THM_OFF 0                // 256 floats: interleaved {theta, mask}
#define Q_OFF   256              // 4 waves * 16*QSTRIDE
#define D_OFF   (Q_OFF + 4 * 16 * QSTRIDE)
#define LDS_FLOATS (D_OFF + 4 * 16 * DSTRIDE)   // 13824 floats = 55296 B

__global__ __launch_bounds__(128)
void masif_geoconv_kernel(const float* __restrict__ rho_coords,
                          const float* __restrict__ theta_coords,
                          const float* __restrict__ input_feat,
                          const float* __restrict__ mask,
                          const float* __restrict__ mu_rho,
                          const float* __restrict__ sigma_rho,
                          const float* __restrict__ mu_theta,
                          const float* __restrict__ sigma_theta,
                          const float* __restrict__ W_conv,
                          const float* __restrict__ b_conv,
                          float* __restrict__ out)
{
    __shared__ float lds[LDS_FLOATS];

    const int n    = blockIdx.x;
    const int tid  = threadIdx.x;
    const int wave = tid >> 5;       // wave w handles feature i = w
    const int lane = tid & 31;
    const int half = lane >> 4;      // 0: lanes 0-15, 1: lanes 16-31
    const int tloc = lane & 15;
    const int i    = wave;

    // ---- stage interleaved {theta, mask} (coalesced, 128 threads, 1 vertex each)
    {
        float tv = theta_coords[n * NV + tid];
        float mv = mask[n * NV + tid];
        lds[THM_OFF + 2 * tid]     = tv;
        lds[THM_OFF + 2 * tid + 1] = mv;
    }

    // ---- per-wave uniform Gaussian parameters (grid structure: r = b/16, t = b%16)
    float mu_r[NR], is2_r[NR];
#pragma unroll
    for (int r = 0; r < NR; ++r) {
        float m = mu_rho[i * NB + r * NT];
        float s = sigma_rho[i * NB + r * NT];
        mu_r[r]  = m;
        is2_r[r] = 1.0f / (s * s + EPSF);
    }
    // per-lane theta params: this lane's B-column t == tloc
    float mu_t, is2_t;
    {
        mu_t = mu_theta[i * NB + tloc];
        float s = sigma_theta[i * NB + tloc];
        is2_t = 1.0f / (s * s + EPSF);
    }

    // ---- build Q in LDS: Q[c][v], c = 2r+{0:feat-weighted,1:plain}, rows 10..15 zero
    float* Q = &lds[Q_OFF + wave * 16 * QSTRIDE];
    for (int v = lane; v < NV; v += 32) {
        float rv = rho_coords[n * NV + v];
        float fv = input_feat[(n * NV + v) * NF + i];
#pragma unroll
        for (int r = 0; r < NR; ++r) {
            float d  = rv - mu_r[r];
            float rg = __expf(-d * d * is2_r[r]);
            Q[(2 * r)     * QSTRIDE + v] = rg * fv;
            Q[(2 * r + 1) * QSTRIDE + v] = rg;
        }
#pragma unroll
        for (int c = 2 * NR; c < 16; ++c) Q[c * QSTRIDE + v] = 0.0f;
    }
    __syncthreads();

    float* Dsc       = &lds[D_OFF + wave * 16 * DSTRIDE];
    const float* thm = &lds[THM_OFF];

    // ---- rotation loop: D = Q^T P via 32 WMMA f32 16x16x4 steps (K = 128)
    for (int k = 0; k < NROT; ++k) {
        const float kofs = (float)k * ROT_STEP;
        v8f acc = {};
        for (int kb = 0; kb < 32; ++kb) {
            const int v0 = 4 * kb + 2 * half;
            // A operand: A[M=tloc, K=v0..v0+1] = Q[v][c=tloc]   (8B-aligned LDS b64)
            v2f a = *(const v2f*)&Q[tloc * QSTRIDE + v0];
            // B operand: B[K=v0+j, N=tloc] = P[v0+j, tloc], computed in-register
            v2f b;
#pragma unroll
            for (int j = 0; j < 2; ++j) {
                float tv = thm[2 * (v0 + j)];
                float mv = thm[2 * (v0 + j) + 1];
                float th = tv + kofs;
                th = (th >= TWO_PI_F) ? th - TWO_PI_F : th;   // mod 2pi (inputs < 4pi)
                float d = th - mu_t;
                b[j] = __expf(-d * d * is2_t) * mv;
            }
            acc = __builtin_amdgcn_wmma_f32_16x16x4_f32(
                false, a, false, b, (short)0, acc, false, false);
        }
        // desc[b = r*16 + t] = num / (den + eps); rows 0..7 -> lanes<16, rows 8..9 -> lanes>=16
        if (half == 0) {
#pragma unroll
            for (int r = 0; r < 4; ++r) {
                Dsc[k * DSTRIDE + r * NT + tloc] = acc[2 * r] / (acc[2 * r + 1] + EPSF);
            }
        } else {
            Dsc[k * DSTRIDE + 4 * NT + tloc] = acc[0] / (acc[1] + EPSF);
        }
    }
    __syncthreads();

    // ---- conv: Desc(16 x 80) @ W_i(80 x 80) + b_i, then max over rotations, relu
    const float* Wi = W_conv + i * NB * NB;
    for (int bt = 0; bt < 5; ++bt) {
        const int col = bt * NT + tloc;
        const float bias = b_conv[i * NB + col];
        v8f acc = {bias, bias, bias, bias, bias, bias, bias, bias};
        for (int kb = 0; kb < 20; ++kb) {
            const int krow = 4 * kb + 2 * half;
            // A: A[M=tloc(rotation), K=krow..krow+1] from Desc LDS (8B aligned)
            v2f a = *(const v2f*)&Dsc[tloc * DSTRIDE + krow];
            // B: W_i[K=krow+j, N=col]
            v2f b;
            b.x = Wi[krow * NB + col];
            b.y = Wi[(krow + 1) * NB + col];
            acc = __builtin_amdgcn_wmma_f32_16x16x4_f32(
                false, a, false, b, (short)0, acc, false, false);
        }
        // C rows: lanes<16 hold M=0..7, lanes>=16 hold M=8..15 (same N=col)
        float m = acc[0];
#pragma unroll
        for (int j = 1; j < 8; ++j) m = fmaxf(m, acc[j]);
        m = fmaxf(m, __shfl_xor(m, 16, 32));   // combine the two row-halves
        if (half == 0) {
            out[(n * NF + i) * NB + col] = fmaxf(m, 0.0f);   // relu(max_k conv)
        }
    }
}

extern "C" void kernel_launch(void* const* d_in, const int* in_sizes, int n_in,
                              void* d_out, int out_size, void* d_ws, size_t ws_size,
                              hipStream_t stream) {
    const float* rho_coords  = (const float*)d_in[0];
    const float* theta_coord = (const float*)d_in[1];
    const float* input_feat  = (const float*)d_in[2];
    const float* mask        = (const float*)d_in[3];
    const float* mu_rho      = (const float*)d_in[4];
    const float* sigma_rho   = (const float*)d_in[5];
    const float* mu_theta    = (const float*)d_in[6];
    const float* sigma_theta = (const float*)d_in[7];
    const float* W_conv      = (const float*)d_in[8];
    const float* b_conv      = (const float*)d_in[9];
    float* out = (float*)d_out;

    const int n_samples = in_sizes[0] / NV;   // 4096
    masif_geoconv_kernel<<<n_samples, 128, 0, stream>>>(
        rho_coords, theta_coord, input_feat, mask,
        mu_rho, sigma_rho, mu_theta, sigma_theta, W_conv, b_conv, out);
}